// VARGP_36404142801181
// MI455X (gfx1250) — compile-verified
//
#include <hip/hip_runtime.h>

// ---------------------------------------------------------------------------
// Problem constants (fixed by the reference)
// ---------------------------------------------------------------------------
constexpr int cH  = 4;      // hyper heads
constexpr int cO  = 64;     // outputs
constexpr int cM  = 64;     // inducing pts per task
constexpr int cT  = 3;      // previous tasks
constexpr int cD  = 128;    // input dim
constexpr int cB  = 512;    // batch
constexpr int cMT = 256;    // total inducing rows = (T+1)*M
constexpr int cNT = 2080;   // tril count M*(M+1)/2
constexpr int cHO = cH * cO;
constexpr float cJIT = 1e-5f;

typedef __attribute__((ext_vector_type(2))) float v2f;
typedef __attribute__((ext_vector_type(8))) float v8f;

// ---------------------------------------------------------------------------
// Hyper prep: inv lengthscales and signal variance
// ---------------------------------------------------------------------------
__global__ void k_hyper(const float* __restrict__ theta, float* __restrict__ ls,
                        float* __restrict__ sf2) {
  int i = blockIdx.x * blockDim.x + threadIdx.x;
  if (i < cH * cD) {
    int h = i / cD, d = i % cD;
    ls[i] = expf(-theta[h * (cD + 1) + d]);
  }
  if (i < cH) sf2[i] = expf(theta[i * (cD + 1) + cD]);
}

// Scaled z_cat rows (prev_z[0..2] then z) + squared norms. 1 thread / (ho,r).
__global__ void k_zs(const float* __restrict__ z, const float* __restrict__ prev_z,
                     const float* __restrict__ ls, float* __restrict__ zs,
                     float* __restrict__ zn) {
  int idx = blockIdx.x * blockDim.x + threadIdx.x;   // ho*cMT + r
  if (idx >= cHO * cMT) return;
  int r = idx & (cMT - 1);
  int ho = idx >> 8;
  int h = ho >> 6, o = ho & 63;
  const float* src;
  if (r < cT * cM) src = prev_z + ((((long)(r >> 6)) * cO + o) * cM + (r & 63)) * cD;
  else             src = z + ((long)o * cM + (r - cT * cM)) * cD;
  const float* l = ls + h * cD;
  float* dst = zs + (long)idx * cD;
  float s = 0.f;
  for (int d = 0; d < cD; ++d) { float v = src[d] * l[d]; dst[d] = v; s += v * v; }
  zn[idx] = s;
}

// Scaled test points per head + norms. 1 thread / (h,b).
__global__ void k_xs(const float* __restrict__ x, const float* __restrict__ ls,
                     float* __restrict__ xs, float* __restrict__ xn) {
  int idx = blockIdx.x * blockDim.x + threadIdx.x;   // h*cB + b
  if (idx >= cH * cB) return;
  int h = idx / cB, b = idx % cB;
  const float* src = x + (long)b * cD;
  const float* l = ls + h * cD;
  float* dst = xs + (long)idx * cD;
  float s = 0.f;
  for (int d = 0; d < cD; ++d) { float v = src[d] * l[d]; dst[d] = v; s += v * v; }
  xn[idx] = s;
}

// ---------------------------------------------------------------------------
// Per-(h,o) WMMA fp32 GEMM: C = op(A) * op(B), templated on transposes so the
// inner K-loop is branch-free. Each wave produces a 16x32 strip of C (two
// 16x16 tiles, dual accumulators) to halve A-fragment traffic and double
// wmma density. K stepped by 4 via V_WMMA_F32_16X16X4_F32.
// Fragment layouts per ISA 7.12.2:
//   A 16x4: lane m = lane&15; VGPRs hold K pair (k,k+1), lane half adds K+2.
//   B/C/D:  row-striped across lanes; rows v / v+8 split by lane half.
// ---------------------------------------------------------------------------
template <bool TA, bool TB>
__global__ void k_gemm(const float* __restrict__ Am, const float* __restrict__ Bm,
                       float* __restrict__ Cm, int Kdim,
                       long sA, long sB, long sC, int lda, int ldb, int ldc,
                       int bGroup) {
  int ho = blockIdx.z;
  const float* Ab = Am + (long)ho * sA;
  const float* Bp = Bm + (long)(ho / bGroup) * sB;
  float* Cb = Cm + (long)ho * sC;
  int lane = threadIdx.x & 31;
  int half = lane >> 4, l16 = lane & 15;
  int tm = blockIdx.y * 16, tn = blockIdx.x * 32;
  int am = tm + l16;
  int bn0 = tn + l16, bn1 = tn + 16 + l16;
  v8f acc0 = {}, acc1 = {};
#pragma unroll 2
  for (int k0 = 0; k0 < Kdim; k0 += 4) {
    int ka = k0 + half * 2;
    v2f a, b0, b1;
    if (!TA) {
      a.x = Ab[(long)am * lda + ka];
      a.y = Ab[(long)am * lda + ka + 1];
    } else {
      a.x = Ab[(long)ka * lda + am];
      a.y = Ab[((long)ka + 1) * lda + am];
    }
    if (!TB) {
      b0.x = Bp[(long)ka * ldb + bn0];
      b0.y = Bp[((long)ka + 1) * ldb + bn0];
      b1.x = Bp[(long)ka * ldb + bn1];
      b1.y = Bp[((long)ka + 1) * ldb + bn1];
    } else {
      b0.x = Bp[(long)bn0 * ldb + ka];
      b0.y = Bp[(long)bn0 * ldb + ka + 1];
      b1.x = Bp[(long)bn1 * ldb + ka];
      b1.y = Bp[(long)bn1 * ldb + ka + 1];
    }
    acc0 = __builtin_amdgcn_wmma_f32_16x16x4_f32(false, a, false, b0, (short)0,
                                                 acc0, false, false);
    acc1 = __builtin_amdgcn_wmma_f32_16x16x4_f32(false, a, false, b1, (short)0,
                                                 acc1, false, false);
  }
#pragma unroll
  for (int v = 0; v < 8; ++v) {
    int row = tm + v + half * 8;
    Cb[(long)row * ldc + tn + l16] = acc0[v];
    Cb[(long)row * ldc + tn + 16 + l16] = acc1[v];
  }
}

// RBF transform of full Gram (in place): K = sf2*exp(-0.5*max(zn_i+zn_j-2G,0))
__global__ void k_rbf_zz(float* __restrict__ Kf, const float* __restrict__ zn,
                         const float* __restrict__ sf2) {
  int idx = blockIdx.x * blockDim.x + threadIdx.x;
  if (idx >= cHO * cMT * cMT) return;
  int ho = idx >> 16;                 // cMT*cMT = 65536
  int rem = idx & 65535;
  int i = rem >> 8, j = rem & 255;
  float g = Kf[idx];
  float d2 = zn[ho * cMT + i] + zn[ho * cMT + j] - 2.f * g;
  Kf[idx] = sf2[ho >> 6] * expf(-0.5f * fmaxf(d2, 0.f));
}

// RBF transform of cross kernel Kzx2 (in place)
__global__ void k_rbf_zx(float* __restrict__ K2, const float* __restrict__ zn,
                         const float* __restrict__ xn, const float* __restrict__ sf2) {
  int idx = blockIdx.x * blockDim.x + threadIdx.x;
  if (idx >= cHO * cMT * cB) return;
  int ho = idx >> 17;                 // cMT*cB = 131072
  int rem = idx & 131071;
  int m = rem >> 9, b = rem & 511;
  int h = ho >> 6;
  float g = K2[idx];
  float d2 = zn[ho * cMT + m] + xn[h * cB + b] - 2.f * g;
  K2[idx] = sf2[h] * expf(-0.5f * fmaxf(d2, 0.f));
}

// V[o] = Lv Lv^T from packed tril vector (row-major tril: idx = i*(i+1)/2 + j)
__global__ void k_V(const float* __restrict__ vec, float* __restrict__ V) {
  int o = blockIdx.x;
  const float* v = vec + (long)o * cNT;
  float* Vo = V + (long)o * cM * cM;
  for (int e = threadIdx.x; e < cM * cM; e += blockDim.x) {
    int i = e >> 6, j = e & 63;
    int n = i < j ? i : j;
    int bi = i * (i + 1) / 2, bj = j * (j + 1) / 2;
    float s = 0.f;
    for (int k = 0; k <= n; ++k) s += v[bi + k] * v[bj + k];
    Vo[e] = s;
  }
}

// S[0:64,0:64] = V (broadcast over h);  mu[0:64] = prev_u_mean[0]
__global__ void k_init(const float* __restrict__ V, const float* __restrict__ pm,
                       float* __restrict__ S, float* __restrict__ mu) {
  int ho = blockIdx.x, o = ho & 63;
  float* Sho = S + (long)ho * cMT * cMT;
  const float* Vo = V + (long)o * cM * cM;
  for (int e = threadIdx.x; e < cM * cM; e += blockDim.x)
    Sho[(e >> 6) * cMT + (e & 63)] = Vo[e];
  float* mh = mu + (long)ho * cMT;
  for (int r = threadIdx.x; r < cM; r += blockDim.x) mh[r] = pm[o * cM + r];
}

// Cholesky of top-left Ml x Ml block of Kf (+jitter) per (h,o), in LDS.
__global__ void k_chol(const float* __restrict__ Kf, float* __restrict__ L, int Ml) {
  extern __shared__ float A[];
  int ho = blockIdx.x;
  const float* src = Kf + (long)ho * cMT * cMT;
  float* dst = L + (long)ho * cMT * cMT;
  for (int e = threadIdx.x; e < Ml * Ml; e += blockDim.x) {
    int i = e / Ml, j = e - i * Ml;
    float v = src[i * cMT + j];
    if (i == j) v += cJIT;
    A[e] = v;
  }
  __syncthreads();
  for (int k = 0; k < Ml; ++k) {
    if (threadIdx.x == 0) A[k * Ml + k] = sqrtf(A[k * Ml + k]);
    __syncthreads();
    float lkk = A[k * Ml + k];
    for (int i = k + 1 + threadIdx.x; i < Ml; i += blockDim.x) A[i * Ml + k] /= lkk;
    __syncthreads();
    for (int i = k + 1 + threadIdx.x; i < Ml; i += blockDim.x) {
      float lik = A[i * Ml + k];
      for (int j = k + 1; j <= i; ++j) A[i * Ml + j] -= lik * A[j * Ml + k];
    }
    __syncthreads();
  }
  for (int e = threadIdx.x; e < Ml * Ml; e += blockDim.x) {
    int i = e / Ml, j = e - i * Ml;
    dst[i * cMT + j] = (j <= i) ? A[e] : 0.f;
  }
}

// Forward (L y = rhs) + backward (L^T x = y) solve, column-per-thread.
// Column state lives in LDS (bank-friendly, thread-private, no barriers).
// Optional sum(alpha^2) per column after the forward pass (for f_var).
__global__ void k_solve(const float* __restrict__ L, const float* __restrict__ RHS,
                        float* __restrict__ OUT, float* __restrict__ sumsq,
                        int Ml, int nrhs, int ldr, int ldo, long sR, long sO,
                        int cols) {
  extern __shared__ float X[];                     // Ml * cols
  int tc = threadIdx.x;
  int col = blockIdx.x * cols + tc;
  if (tc >= cols || col >= nrhs) return;
  int ho = blockIdx.y;
  const float* Lho = L + (long)ho * cMT * cMT;
  const float* R = RHS + (long)ho * sR;
  float* Oo = OUT + (long)ho * sO;
  for (int i = 0; i < Ml; ++i) {
    float xv = R[(long)i * ldr + col];
    for (int k = 0; k < i; ++k) xv -= Lho[i * cMT + k] * X[k * cols + tc];
    X[i * cols + tc] = xv / Lho[i * cMT + i];
  }
  if (sumsq) {
    float s = 0.f;
    for (int i = 0; i < Ml; ++i) { float v = X[i * cols + tc]; s += v * v; }
    sumsq[(long)ho * nrhs + col] = s;
  }
  for (int i = Ml - 1; i >= 0; --i) {
    float xv = X[i * cols + tc];
    for (int k = i + 1; k < Ml; ++k) xv -= Lho[k * cMT + i] * X[k * cols + tc];
    xv /= Lho[i * cMT + i];
    X[i * cols + tc] = xv;
    Oo[(long)i * ldo + col] = xv;
  }
}

// mu[Ml+r] = beta(:,r) . mu[0:Ml] + b[r]
__global__ void k_mut(const float* __restrict__ beta, float* __restrict__ mu,
                      const float* __restrict__ bv, int Ml) {
  int idx = blockIdx.x * blockDim.x + threadIdx.x;  // ho*64 + r
  if (idx >= cHO * cM) return;
  int ho = idx >> 6, r = idx & 63, o = ho & 63;
  const float* bb = beta + (long)ho * cMT * cM;
  float* m = mu + (long)ho * cMT;
  float s = bv[o * cM + r];
  for (int i = 0; i < Ml; ++i) s += bb[i * cM + r] * m[i];
  m[Ml + r] = s;
}

// Joint update: S[:Ml,Ml:] = S_AT, symmetric mirror, S[Ml:,Ml:] = cov + V
__global__ void k_upS(const float* __restrict__ SAT, const float* __restrict__ cov,
                      const float* __restrict__ V, float* __restrict__ S, int Ml) {
  int ho = blockIdx.x, o = ho & 63;
  float* Sho = S + (long)ho * cMT * cMT;
  const float* sat = SAT + (long)ho * cMT * cM;
  const float* cv = cov + (long)ho * cM * cM;
  const float* Vo = V + (long)o * cM * cM;
  for (int e = threadIdx.x; e < Ml * cM; e += blockDim.x) {
    int i = e >> 6, r = e & 63;
    float v = sat[e];
    Sho[i * cMT + (Ml + r)] = v;
    Sho[(Ml + r) * cMT + i] = v;
  }
  for (int e = threadIdx.x; e < cM * cM; e += blockDim.x) {
    int i = e >> 6, j = e & 63;
    Sho[(Ml + i) * cMT + (Ml + j)] = cv[e] + Vo[e];
  }
}

// Fused WMMA: fvar3[ho,b] = sum_m beta[m,b] * (S @ beta)[m,b]
// One wave per 16-column strip; loops all 16 row-tiles, accumulating column
// sums in registers; lane halves combined through a tiny LDS buffer.
__global__ void k_sbeta(const float* __restrict__ S, const float* __restrict__ beta,
                        float* __restrict__ fvar3) {
  __shared__ float tmp[32];
  int ho = blockIdx.y;
  const float* Sho = S + (long)ho * cMT * cMT;
  const float* Bho = beta + (long)ho * cMT * cB;
  int lane = threadIdx.x & 31;
  int half = lane >> 4, l16 = lane & 15;
  int tn = blockIdx.x * 16;
  float colsum = 0.f;
  for (int tm = 0; tm < cMT; tm += 16) {
    v8f acc = {};
#pragma unroll 4
    for (int k0 = 0; k0 < cMT; k0 += 4) {
      int ka = k0 + half * 2;
      v2f a, b;
      a.x = Sho[(long)(tm + l16) * cMT + ka];
      a.y = Sho[(long)(tm + l16) * cMT + ka + 1];
      b.x = Bho[(long)ka * cB + tn + l16];
      b.y = Bho[((long)ka + 1) * cB + tn + l16];
      acc = __builtin_amdgcn_wmma_f32_16x16x4_f32(false, a, false, b, (short)0, acc,
                                                  false, false);
    }
#pragma unroll
    for (int v = 0; v < 8; ++v) {
      int row = tm + v + half * 8;
      colsum += acc[v] * Bho[(long)row * cB + tn + l16];
    }
  }
  tmp[lane] = colsum;
  __syncthreads();
  if (lane < 16) fvar3[(long)ho * cB + tn + lane] = tmp[lane] + tmp[lane + 16];
}

// Outputs: pred_mu = beta^T mu ; pred_var = sf2 - sum(alpha^2) + beta.(S beta)
__global__ void k_out(const float* __restrict__ beta, const float* __restrict__ mu,
                      const float* __restrict__ ssq, const float* __restrict__ fv3,
                      const float* __restrict__ sf2, float* __restrict__ out) {
  int idx = blockIdx.x * blockDim.x + threadIdx.x;   // ho*cB + b
  if (idx >= cHO * cB) return;
  int ho = idx >> 9, b = idx & 511, h = ho >> 6;
  const float* Bho = beta + (long)ho * cMT * cB;
  const float* m = mu + (long)ho * cMT;
  float s = 0.f;
  for (int i = 0; i < cMT; ++i) s += Bho[(long)i * cB + b] * m[i];
  out[idx] = s;
  out[(long)cHO * cB + idx] = sf2[h] - ssq[idx] + fv3[idx];
}

// ---------------------------------------------------------------------------
extern "C" void kernel_launch(void* const* d_in, const int* in_sizes, int n_in,
                              void* d_out, int out_size, void* d_ws, size_t ws_size,
                              hipStream_t stream) {
  const float* x      = (const float*)d_in[0];
  const float* theta  = (const float*)d_in[1];
  const float* z      = (const float*)d_in[2];
  const float* u_mean = (const float*)d_in[3];
  const float* u_tril = (const float*)d_in[4];
  const float* pz     = (const float*)d_in[5];
  const float* pum    = (const float*)d_in[6];
  const float* put    = (const float*)d_in[7];

  // ---- workspace layout (floats), ~412 MB total ----
  float* w = (float*)d_ws;
  size_t off = 0;
  auto alloc = [&](size_t n) { float* p = w + off; off += (n + 63) & ~(size_t)63; return p; };
  float* ls  = alloc((size_t)cH * cD);
  float* sf2 = alloc(cH);
  float* zs  = alloc((size_t)cHO * cMT * cD);
  float* zn  = alloc((size_t)cHO * cMT);
  float* xs  = alloc((size_t)cH * cB * cD);
  float* xn  = alloc((size_t)cH * cB);
  float* Kf  = alloc((size_t)cHO * cMT * cMT);
  float* Lm  = alloc((size_t)cHO * cMT * cMT);
  float* S   = alloc((size_t)cHO * cMT * cMT);
  float* mu  = alloc((size_t)cHO * cMT);
  float* bet = alloc((size_t)cHO * cMT * cM);
  float* SAT = alloc((size_t)cHO * cMT * cM);
  float* cov = alloc((size_t)cHO * cM * cM);
  float* Vb  = alloc((size_t)cO * cM * cM);
  float* K2  = alloc((size_t)cHO * cMT * cB);
  float* ssq = alloc((size_t)cHO * cB);
  float* fv3 = alloc((size_t)cHO * cB);
  if (ws_size < off * sizeof(float)) return;  // scratch too small: bail

  // allow >64KB dynamic LDS for Cholesky / triangular solve kernels
  (void)hipFuncSetAttribute((const void*)k_chol,
                            hipFuncAttributeMaxDynamicSharedMemorySize, 262144);
  (void)hipFuncSetAttribute((const void*)k_solve,
                            hipFuncAttributeMaxDynamicSharedMemorySize, 131072);

  // 1) hypers / scaled points / norms
  k_hyper<<<dim3((cH * cD + 255) / 256), 256, 0, stream>>>(theta, ls, sf2);
  k_zs<<<dim3(cHO * cMT / 256), 256, 0, stream>>>(z, pz, ls, zs, zn);
  k_xs<<<dim3(cH * cB / 256), 256, 0, stream>>>(x, ls, xs, xn);

  // 2) full Gram (WMMA, B = zs^T) + RBF transform
  k_gemm<false, true><<<dim3(cMT / 32, cMT / 16, cHO), 32, 0, stream>>>(
      zs, zs, Kf, cD, (long)cMT * cD, (long)cMT * cD, (long)cMT * cMT,
      cD, cD, cMT, 1);
  k_rbf_zz<<<dim3(cHO * cMT * cMT / 256), 256, 0, stream>>>(Kf, zn, sf2);

  // 3) init S, mu from task 0
  k_V<<<dim3(cO), 256, 0, stream>>>(put, Vb);
  k_init<<<dim3(cHO), 256, 0, stream>>>(Vb, pum, S, mu);

  // 4) recursive linear_joint steps (Ml = 64, 128, 192)
  for (int t = 1; t <= cT; ++t) {
    int Ml = t * cM;
    const float* bv = (t < cT) ? pum + (size_t)t * cO * cM : u_mean;
    const float* tv = (t < cT) ? put + (size_t)t * cO * cNT : u_tril;
    k_chol<<<dim3(cHO), 256, Ml * Ml * 4, stream>>>(Kf, Lm, Ml);
    // RHS = Kzx = Kf[:, Ml:Ml+64]; OUT = beta (Kzz^{-1} Kzx), ld 64
    k_solve<<<dim3(1, cHO), 64, Ml * cM * 4, stream>>>(
        Lm, Kf + Ml, bet, nullptr, Ml, cM, cMT, cM,
        (long)cMT * cMT, (long)cMT * cM, cM);
    k_mut<<<dim3(cHO * cM / 256), 256, 0, stream>>>(bet, mu, bv, Ml);
    // S_AT = S @ beta (WMMA)
    k_gemm<false, false><<<dim3(cM / 32, Ml / 16, cHO), 32, 0, stream>>>(
        S, bet, SAT, Ml, (long)cMT * cMT, (long)cMT * cM, (long)cMT * cM,
        cMT, cM, cM, 1);
    // cov_tt = beta^T @ S_AT (WMMA, transA)
    k_gemm<true, false><<<dim3(cM / 32, cM / 16, cHO), 32, 0, stream>>>(
        bet, SAT, cov, Ml, (long)cMT * cM, (long)cMT * cM,
        (long)cM * cM, cM, cM, cM, 1);
    k_V<<<dim3(cO), 256, 0, stream>>>(tv, Vb);
    k_upS<<<dim3(cHO), 256, 0, stream>>>(SAT, cov, Vb, S, Ml);
  }

  // 5) final marginal: Kzx2 (WMMA) + RBF, chol(256), solves, fused S@beta
  k_gemm<false, true><<<dim3(cB / 32, cMT / 16, cHO), 32, 0, stream>>>(
      zs, xs, K2, cD, (long)cMT * cD, (long)cB * cD, (long)cMT * cB,
      cD, cD, cB, cO);
  k_rbf_zx<<<dim3(cHO * cMT * cB / 256), 256, 0, stream>>>(K2, zn, xn, sf2);
  k_chol<<<dim3(cHO), 256, cMT * cMT * 4, stream>>>(Kf, Lm, cMT);
  // alpha^2 column sums computed inline; beta overwrites K2 in place
  k_solve<<<dim3(cB / 128, cHO), 128, cMT * 128 * 4, stream>>>(
      Lm, K2, K2, ssq, cMT, cB, cB, cB, (long)cMT * cB, (long)cMT * cB, 128);
  k_sbeta<<<dim3(cB / 16, cHO), 32, 0, stream>>>(S, K2, fv3);
  k_out<<<dim3(cHO * cB / 256), 256, 0, stream>>>(K2, mu, ssq, fv3, sf2,
                                                  (float*)d_out);
}